// PointConvNet_71030169141849
// MI455X (gfx1250) — compile-verified
//
#include <hip/hip_runtime.h>
#include <hip/hip_bf16.h>

typedef __attribute__((ext_vector_type(2))) float v2f;
typedef __attribute__((ext_vector_type(8))) float v8f;

#define D        64
#define KPAD     68          // 64 x-feats + 3 rel-pos + 1 homogeneous (bias) coord
#define NPAIR    34          // 68 K-values as 34 (k,k+1) pairs
#define WAVES    2
#define BLOCK    (WAVES * 32)
#define TILE_M   32          // two 16-row WMMA tiles per wave -> B-frag reuse x2

__device__ __forceinline__ float neg_inf() {
  return __uint_as_float(0xff800000u);
}

// Initialize output: segment-max identity (-inf) for the feature block,
// passthrough copies of pos (float) and batch (int bits) for the tuple tail.
__global__ void pc_init_out(float* __restrict__ out,
                            const float* __restrict__ pos,
                            const int* __restrict__ batch,
                            int n_nodes) {
  const int i = blockIdx.x * blockDim.x + threadIdx.x;
  const int nOut = n_nodes * D;
  const int nPos = n_nodes * 3;
  if (i < nOut) {
    out[i] = neg_inf();
  } else if (i < nOut + nPos) {
    out[i] = pos[i - nOut];
  } else if (i < nOut + nPos + n_nodes) {
    ((int*)out)[i] = batch[i - nOut - nPos];
  }
}

__global__ __launch_bounds__(BLOCK)
void pc_edge_kernel(const float* __restrict__ x,
                    const float* __restrict__ pos,
                    const int* __restrict__ ei,
                    const float* __restrict__ W1,
                    const float* __restrict__ b1,
                    const float* __restrict__ W2,
                    const float* __restrict__ b2,
                    float* __restrict__ out,
                    int n_nodes, int n_edges)
{
  // Pair-swizzled weights: sWp[(t*64 + n)*2 + v] = W[2t+v][n]  -> one ds_load_b64
  // per B fragment, no repacking moves. Bias folded in as extra K rows:
  //   W1 row 67 = b1 (feat col 67 = 1.0) ; W2 row 64 = b2, rows 65..67 = 0.
  __shared__ __align__(16) float sW1p[NPAIR * D * 2];
  __shared__ __align__(16) float sW2p[NPAIR * D * 2];
  __shared__ __align__(16) float sFeat[WAVES][TILE_M * KPAD];
  __shared__ int   sDst[WAVES][TILE_M];

  const int tid = threadIdx.x;

  // ---- block-cooperative weight preload (once per block) ----
  for (int i = tid; i < NPAIR * D; i += BLOCK) {
    const int t = i >> 6;          // k-pair index, k0 = 2t, k1 = 2t+1
    const int n = i & 63;
    const int k0 = 2 * t, k1 = k0 + 1;
    sW1p[2 * i]     = W1[k0 * D + n];                      // k0 <= 66 < 67 always
    sW1p[2 * i + 1] = (k1 < D + 3) ? W1[k1 * D + n] : b1[n];
    sW2p[2 * i]     = (k0 < D) ? W2[k0 * D + n] : ((k0 == D) ? b2[n] : 0.0f);
    sW2p[2 * i + 1] = (k1 < D) ? W2[k1 * D + n] : 0.0f;    // k1 odd, never == 64
  }
  __syncthreads();

  const int lane = tid & 31;
  const int w    = tid >> 5;
  const int lo   = lane & 15;   // N / M index within half-wave
  const int hi   = lane >> 4;   // half-wave select (K-pair select in A/B frags)

  const int e_total = n_edges + n_nodes;            // edges + self loops
  const int n_tiles = (e_total + TILE_M - 1) / TILE_M;
  const int wave0   = blockIdx.x * WAVES + w;
  const int wstride = gridDim.x * WAVES;

  const int* __restrict__ srcIdx = ei;              // edge_index[0] = j (source)
  const int* __restrict__ dstIdx = ei + n_edges;    // edge_index[1] = i (target)
  float* sF = &sFeat[w][0];

  for (int t = wave0; t < n_tiles; t += wstride) {
    const int ebase = t * TILE_M;

    // ---- stage 32x68 feature tile, one lane per edge row ----
    {
      const int m = lane;
      const int e = ebase + m;
      int s = 0, dd = 0;
      const bool valid = (e < e_total);
      if (valid) {
        if (e < n_edges) { s = srcIdx[e]; dd = dstIdx[e]; }
        else             { s = e - n_edges; dd = s; }   // self loop
      }
      const float4* xr = (const float4*)(x + (size_t)s * D);
      float4* fr = (float4*)(sF + m * KPAD);
#pragma unroll
      for (int j = 0; j < 16; ++j) fr[j] = xr[j];
      float4 tail;                                  // [rel | 1.0 homogeneous]
      tail.x = pos[s * 3 + 0] - pos[dd * 3 + 0];
      tail.y = pos[s * 3 + 1] - pos[dd * 3 + 1];
      tail.z = pos[s * 3 + 2] - pos[dd * 3 + 2];
      tail.w = 1.0f;
      fr[16] = tail;
      sDst[w][m] = valid ? dd : -1;
    }

    // ---- GEMM1: h = relu([feat|1] @ [W1;b1]), (32x68)@(68x64) fp32 WMMA ----
    v8f acc[2][4];
#pragma unroll
    for (int u = 0; u < 2; ++u)
#pragma unroll
      for (int nt = 0; nt < 4; ++nt)
        acc[u][nt] = (v8f){0, 0, 0, 0, 0, 0, 0, 0};

#pragma unroll
    for (int kc = 0; kc < KPAD / 4; ++kc) {
      const int ka = kc * 4 + 2 * hi;
      const v2f a0 = *(const v2f*)(sF + lo * KPAD + ka);
      const v2f a1 = *(const v2f*)(sF + (16 + lo) * KPAD + ka);
      const float* bp = sW1p + ((2 * kc + hi) * D + lo) * 2;
#pragma unroll
      for (int nt = 0; nt < 4; ++nt) {
        const v2f b = *(const v2f*)(bp + nt * 32);
        acc[0][nt] = __builtin_amdgcn_wmma_f32_16x16x4_f32(
            false, a0, false, b, (short)0, acc[0][nt], false, false);
        acc[1][nt] = __builtin_amdgcn_wmma_f32_16x16x4_f32(
            false, a1, false, b, (short)0, acc[1][nt], false, false);
      }
    }

    // ---- ReLU; write h back (C/D layout -> row-major LDS, reuse tile) ----
#pragma unroll
    for (int u = 0; u < 2; ++u)
#pragma unroll
      for (int nt = 0; nt < 4; ++nt)
#pragma unroll
        for (int r = 0; r < 8; ++r) {
          float v = acc[u][nt][r];
          v = v > 0.0f ? v : 0.0f;
          sF[(u * 16 + r + 8 * hi) * KPAD + lo + nt * 16] = v;
        }
    {   // homogeneous tail for GEMM2: h[m][64..67] = {1,0,0,0}, one b128/lane
      float4* hp = (float4*)(sF + lane * KPAD + D);
      float4 one; one.x = 1.0f; one.y = 0.0f; one.z = 0.0f; one.w = 0.0f;
      *hp = one;
    }

    // ---- GEMM2: msg = [h|1] @ [W2;b2], (32x68)@(68x64) ----
#pragma unroll
    for (int u = 0; u < 2; ++u)
#pragma unroll
      for (int nt = 0; nt < 4; ++nt)
        acc[u][nt] = (v8f){0, 0, 0, 0, 0, 0, 0, 0};

#pragma unroll
    for (int kc = 0; kc < KPAD / 4; ++kc) {
      const int ka = kc * 4 + 2 * hi;
      const v2f a0 = *(const v2f*)(sF + lo * KPAD + ka);
      const v2f a1 = *(const v2f*)(sF + (16 + lo) * KPAD + ka);
      const float* bp = sW2p + ((2 * kc + hi) * D + lo) * 2;
#pragma unroll
      for (int nt = 0; nt < 4; ++nt) {
        const v2f b = *(const v2f*)(bp + nt * 32);
        acc[0][nt] = __builtin_amdgcn_wmma_f32_16x16x4_f32(
            false, a0, false, b, (short)0, acc[0][nt], false, false);
        acc[1][nt] = __builtin_amdgcn_wmma_f32_16x16x4_f32(
            false, a1, false, b, (short)0, acc[1][nt], false, false);
      }
    }

    // ---- scatter-max: one f32 max-atomic per output element ----
#pragma unroll
    for (int u = 0; u < 2; ++u)
#pragma unroll
      for (int nt = 0; nt < 4; ++nt)
#pragma unroll
        for (int r = 0; r < 8; ++r) {
          const int m  = u * 16 + r + 8 * hi;
          const int dd = sDst[w][m];
          if (dd >= 0) {
            float* p = out + (size_t)dd * D + lo + nt * 16;
            (void)__hip_atomic_fetch_max(p, acc[u][nt][r],
                                         __ATOMIC_RELAXED, __HIP_MEMORY_SCOPE_AGENT);
          }
        }
  }
}

extern "C" void kernel_launch(void* const* d_in, const int* in_sizes, int n_in,
                              void* d_out, int out_size, void* d_ws, size_t ws_size,
                              hipStream_t stream) {
  const float* x     = (const float*)d_in[0];
  const float* pos   = (const float*)d_in[1];
  const int*   batch = (const int*)d_in[2];
  const int*   ei    = (const int*)d_in[3];
  const float* W1    = (const float*)d_in[4];
  const float* b1    = (const float*)d_in[5];
  const float* W2    = (const float*)d_in[6];
  const float* b2    = (const float*)d_in[7];
  float* out = (float*)d_out;

  const int n_nodes = in_sizes[0] / D;     // x is [N, 64]
  const int n_edges = in_sizes[3] / 2;     // edge_index is [2, E]

  const int total = n_nodes * (D + 3 + 1); // out | pos | batch
  pc_init_out<<<(total + 255) / 256, 256, 0, stream>>>(out, pos, batch, n_nodes);

  const int blocks = 5120;                 // persistent; grid-stride over edge tiles
  pc_edge_kernel<<<blocks, BLOCK, 0, stream>>>(x, pos, ei, W1, b1, W2, b2,
                                               out, n_nodes, n_edges);
}